// PairwissDistanceLoss_47545287967528
// MI455X (gfx1250) — compile-verified
//
#include <hip/hip_runtime.h>
#include <hip/hip_bf16.h>
#include <math.h>

typedef __attribute__((ext_vector_type(2))) float v2f;
typedef __attribute__((ext_vector_type(8))) float v8f;

#define N_ROWS 1024
#define DIMS   8192
#define NT2    32            // N_ROWS / 32 macro-tile blocks (each 2x2 WMMA tiles)
#define NPAIRS 523776.0      // 1024*1023/2

// ---------------------------------------------------------------------------
// Kernel 1: per-row squared norms (Gram diagonal), exact f32 like reference.
// grid = (1024 rows, 2 matrices), block = 256
// ---------------------------------------------------------------------------
__global__ __launch_bounds__(256) void row_sqnorm_kernel(
    const float* __restrict__ pred, const float* __restrict__ target,
    float* __restrict__ sqp, float* __restrict__ sqt)
{
    const int row = blockIdx.x;
    const float* src = (blockIdx.y == 0) ? pred : target;
    float* dst = (blockIdx.y == 0) ? sqp : sqt;
    const float* p = src + (size_t)row * DIMS;

    float s = 0.0f;
    for (int k = threadIdx.x * 4; k < DIMS; k += 256 * 4) {
        float4 v = *(const float4*)(p + k);
        s = fmaf(v.x, v.x, fmaf(v.y, v.y, fmaf(v.z, v.z, fmaf(v.w, v.w, s))));
    }
    __shared__ float red[256];
    red[threadIdx.x] = s;
    __syncthreads();
    for (int w = 128; w >= 1; w >>= 1) {
        if ((int)threadIdx.x < w) red[threadIdx.x] += red[threadIdx.x + w];
        __syncthreads();
    }
    if (threadIdx.x == 0) dst[row] = red[0];
}

// ---------------------------------------------------------------------------
// Per-16x16-subtile fused distance loss. C/D layout: lane holds N = lane&15,
// accumulator element v holds M = v + 8*(lane>=16).
// ---------------------------------------------------------------------------
__device__ __forceinline__ float tile_loss(
    const v8f& aP, const v8f& aT, int ibase, int j, float sqpj, float sqtj,
    const float* __restrict__ sqp, const float* __restrict__ sqt, int half)
{
    float s = 0.0f;
#pragma unroll
    for (int v = 0; v < 8; ++v) {
        const int i = ibase + v + 8 * half;
        if (i < j) {   // upper-triangle pairs only (matches triu_indices k=1)
            float d2p = fmaxf(sqp[i] + sqpj - 2.0f * aP[v], 0.0f);
            float d2t = fmaxf(sqt[i] + sqtj - 2.0f * aT[v], 0.0f);
            float diff = sqrtf(d2p) - sqrtf(d2t);
            s = fmaf(diff, diff, s);
        }
    }
    return s;
}

// ---------------------------------------------------------------------------
// Kernel 2: one wave32 per 32x32 macro-tile = 2x2 register-blocked 16x16
// WMMA tiles. A fragments reused across two B fragments -> 8 loads : 8 WMMAs
// per K-step for pred+target combined. EXEC all-ones at every WMMA.
// grid = (32, 32), block = 32 (1 wave).
// ---------------------------------------------------------------------------
__global__ __launch_bounds__(32) void pair_tile_kernel(
    const float* __restrict__ pred, const float* __restrict__ target,
    const float* __restrict__ sqp, const float* __restrict__ sqt,
    float* __restrict__ partials)
{
    const int mi = blockIdx.x;       // A-side (row) macro tile -> i
    const int mj = blockIdx.y;       // B-side (col) macro tile -> j
    const int pidx = mi * NT2 + mj;
    const int lane = threadIdx.x;    // 0..31 (wave32)

    if (mi > mj) {                   // strictly-lower macro tiles: nothing
        if (lane == 0) partials[pidx] = 0.0f;
        return;
    }

    const int half = lane >> 4;      // 0: K%4 in {0,1}; 1: K%4 in {2,3}
    const int l16  = lane & 15;
    const int ko   = half * 2;

    // 16x4 f32 A layout / 4x16 f32 B layout: both lane halves use row l16
    // of their respective 16-row block.
    const size_t rA0 = (size_t)(mi * 32 + l16) * DIMS;
    const size_t rA1 = rA0 + (size_t)16 * DIMS;
    const size_t rB0 = (size_t)(mj * 32 + l16) * DIMS;
    const size_t rB1 = rB0 + (size_t)16 * DIMS;

    const float* pa0 = pred + rA0;
    const float* pa1 = pred + rA1;
    const float* pb0 = pred + rB0;
    const float* pb1 = pred + rB1;
    const float* ta0 = target + rA0;
    const float* ta1 = target + rA1;
    const float* tb0 = target + rB0;
    const float* tb1 = target + rB1;

    v8f aP00 = {}, aP01 = {}, aP10 = {}, aP11 = {};
    v8f aT00 = {}, aT01 = {}, aT10 = {}, aT11 = {};

#pragma unroll 2
    for (int k = 0; k < DIMS; k += 4) {
        v2f a0 = *(const v2f*)(pa0 + k + ko);
        v2f a1 = *(const v2f*)(pa1 + k + ko);
        v2f b0 = *(const v2f*)(pb0 + k + ko);
        v2f b1 = *(const v2f*)(pb1 + k + ko);
        aP00 = __builtin_amdgcn_wmma_f32_16x16x4_f32(false, a0, false, b0, (short)0, aP00, false, false);
        aP01 = __builtin_amdgcn_wmma_f32_16x16x4_f32(false, a0, false, b1, (short)0, aP01, false, false);
        aP10 = __builtin_amdgcn_wmma_f32_16x16x4_f32(false, a1, false, b0, (short)0, aP10, false, false);
        aP11 = __builtin_amdgcn_wmma_f32_16x16x4_f32(false, a1, false, b1, (short)0, aP11, false, false);

        v2f c0 = *(const v2f*)(ta0 + k + ko);
        v2f c1 = *(const v2f*)(ta1 + k + ko);
        v2f d0 = *(const v2f*)(tb0 + k + ko);
        v2f d1 = *(const v2f*)(tb1 + k + ko);
        aT00 = __builtin_amdgcn_wmma_f32_16x16x4_f32(false, c0, false, d0, (short)0, aT00, false, false);
        aT01 = __builtin_amdgcn_wmma_f32_16x16x4_f32(false, c0, false, d1, (short)0, aT01, false, false);
        aT10 = __builtin_amdgcn_wmma_f32_16x16x4_f32(false, c1, false, d0, (short)0, aT10, false, false);
        aT11 = __builtin_amdgcn_wmma_f32_16x16x4_f32(false, c1, false, d1, (short)0, aT11, false, false);
    }

    // Fused epilogue over the four 16x16 sub-tiles.
    const int j0 = mj * 32 + l16;
    const int j1 = j0 + 16;
    const int i0 = mi * 32;
    const int i1 = i0 + 16;
    const float sqpj0 = sqp[j0], sqtj0 = sqt[j0];
    const float sqpj1 = sqp[j1], sqtj1 = sqt[j1];

    float s = tile_loss(aP00, aT00, i0, j0, sqpj0, sqtj0, sqp, sqt, half)
            + tile_loss(aP01, aT01, i0, j1, sqpj1, sqtj1, sqp, sqt, half)
            + tile_loss(aP10, aT10, i1, j0, sqpj0, sqtj0, sqp, sqt, half)
            + tile_loss(aP11, aT11, i1, j1, sqpj1, sqtj1, sqp, sqt, half);

    // wave32 reduction
#pragma unroll
    for (int off = 16; off >= 1; off >>= 1)
        s += __shfl_xor(s, off, 32);

    if (lane == 0) partials[pidx] = s;
}

// ---------------------------------------------------------------------------
// Kernel 3: sum 1024 macro-tile partials in double, divide by pair count.
// ---------------------------------------------------------------------------
__global__ __launch_bounds__(256) void finalize_kernel(
    const float* __restrict__ partials, float* __restrict__ out)
{
    double s = 0.0;
    for (int i = threadIdx.x; i < NT2 * NT2; i += 256)
        s += (double)partials[i];

    __shared__ double red[256];
    red[threadIdx.x] = s;
    __syncthreads();
    for (int w = 128; w >= 1; w >>= 1) {
        if ((int)threadIdx.x < w) red[threadIdx.x] += red[threadIdx.x + w];
        __syncthreads();
    }
    if (threadIdx.x == 0) out[0] = (float)(red[0] / NPAIRS);
}

// ---------------------------------------------------------------------------
extern "C" void kernel_launch(void* const* d_in, const int* in_sizes, int n_in,
                              void* d_out, int out_size, void* d_ws, size_t ws_size,
                              hipStream_t stream)
{
    const float* pred   = (const float*)d_in[0];
    const float* target = (const float*)d_in[1];
    float* out = (float*)d_out;

    // ws layout: [0..1023] macro-tile partials, [1024..2047] sqp, [2048..3071] sqt
    float* partials = (float*)d_ws;
    float* sqp = partials + NT2 * NT2;
    float* sqt = sqp + N_ROWS;

    row_sqnorm_kernel<<<dim3(N_ROWS, 2), 256, 0, stream>>>(pred, target, sqp, sqt);
    pair_tile_kernel<<<dim3(NT2, NT2), 32, 0, stream>>>(pred, target, sqp, sqt, partials);
    finalize_kernel<<<1, 256, 0, stream>>>(partials, out);
}